// BaseModel_54614804136216
// MI455X (gfx1250) — compile-verified
//
#include <hip/hip_runtime.h>
#include <stdint.h>

// Problem constants (match reference)
#define PNUM  128
#define BATCH 512
#define HID   512
#define ALPHA 0.5f

#define ROWS_PER_CHUNK 16              // gather-mode 16-bit-index limit
#define ROW_BYTES (HID * 4)            // 2048 bytes per particle row

typedef uint32_t v4u __attribute__((ext_vector_type(4)));
typedef uint32_t v8u __attribute__((ext_vector_type(8)));

__device__ __forceinline__ uint32_t lds_addr_of(const void* p) {
  // Generic pointers into the LDS aperture carry the wave-relative LDS byte
  // offset in their low 32 bits (aperture base has zero low bits).
  return (uint32_t)(uintptr_t)p;
}

__device__ __forceinline__ uint32_t hash32(uint32_t x) {
  x ^= x >> 16; x *= 0x7feb352du;
  x ^= x >> 15; x *= 0x846ca68bu;
  x ^= x >> 16;
  return x;
}

// ---------------------------------------------------------------------------
// Kernel 1: per-batch-element soft-resampling bookkeeping (unchanged).
// grid = BATCH blocks, block = PNUM (=128) threads.
// ---------------------------------------------------------------------------
__global__ void resample_kernel(const float* __restrict__ prob,
                                int* __restrict__ srcrow,
                                float* __restrict__ prob_out) {
  const int b = blockIdx.x;
  const int p = threadIdx.x;
  __shared__ float cdf[PNUM];
  __shared__ float red[PNUM];

  const float logw = prob[p * BATCH + b];
  const float w = __expf(logw);
  const float q = ALPHA * w + (1.0f - ALPHA) / (float)PNUM;

  // Inclusive prefix sum (Hillis-Steele) over the particle axis.
  cdf[p] = q;
  __syncthreads();
  for (int off = 1; off < PNUM; off <<= 1) {
    float v = (p >= off) ? cdf[p - off] : 0.0f;
    __syncthreads();
    cdf[p] += v;
    __syncthreads();
  }
  const float total = cdf[PNUM - 1];

  // Deterministic counter-based uniform for draw (p, b).
  uint32_t h = hash32((uint32_t)(b * PNUM + p) * 2654435761u + 0x9E3779B9u);
  const float u = (float)(h >> 8) * (1.0f / 16777216.0f) * total;

  // lower_bound: smallest idx with cdf[idx] >= u (7-step binary search).
  int idx = 0;
  #pragma unroll
  for (int s = PNUM >> 1; s >= 1; s >>= 1) {
    int n = idx + s;
    if (n < PNUM && cdf[n - 1] < u) idx = n;
  }

  // Flat layout: output row (p,b) = p*BATCH + b, source row = idx*BATCH + b.
  srcrow[p * BATCH + b] = idx * BATCH + b;

  // Reweight: w' = w / (alpha*w + (1-alpha)/P), then log-softmax over p.
  const float lw_src = prob[idx * BATCH + b];
  const float ws = __expf(lw_src);
  const float wp = ws / (ALPHA * ws + (1.0f - ALPHA) / (float)PNUM);
  const float lwp = __logf(wp);

  red[p] = lwp;
  __syncthreads();
  for (int off = PNUM >> 1; off >= 1; off >>= 1) {
    if (p < off) red[p] = fmaxf(red[p], red[p + off]);
    __syncthreads();
  }
  const float m = red[0];
  __syncthreads();
  red[p] = __expf(lwp - m);
  __syncthreads();
  for (int off = PNUM >> 1; off >= 1; off >>= 1) {
    if (p < off) red[p] += red[p + off];
    __syncthreads();
  }
  const float lse = m + __logf(red[0]);
  prob_out[p * BATCH + b] = lwp - lse;
}

// ---------------------------------------------------------------------------
// Kernel 2: TDM gather. One wave per block; each block moves 16 rows (32 KB)
// with a single gather-mode tensor_load_to_lds (16-bit row indices) plus one
// normal-mode tensor_store_from_lds. Tensor ops complete in order per wave
// and are tracked by TENSORcnt, so one wait separates load and store.
// ---------------------------------------------------------------------------
__global__ void gather_tdm_kernel(const float* __restrict__ particles,
                                  const int* __restrict__ srcrow,
                                  float* __restrict__ out) {
  __shared__ float buf[ROWS_PER_CHUNK * HID];   // 32 KB tile staging

  const int row0 = blockIdx.x * ROWS_PER_CHUNK;          // wave-uniform
  const uint32_t lds = lds_addr_of(&buf[0]);

  // 16 gather row indices (uniform scalar loads; values < 65536 fit u16).
  uint32_t ridx[ROWS_PER_CHUNK];
  #pragma unroll
  for (int i = 0; i < ROWS_PER_CHUNK; ++i)
    ridx[i] = (uint32_t)srcrow[row0 + i];

  // ---- D# for the gather load -------------------------------------------
  const uint64_t gp = (uint64_t)(uintptr_t)particles;
  v4u g0;
  g0.x = 0x80000001u;                       // gather_mode=1, 16b idx, count=1
  g0.y = lds;                               // lds_addr
  g0.z = (uint32_t)gp;                      // global_addr[31:0]
  g0.w = ((uint32_t)(gp >> 32) & 0x1FFFFFFu) | (2u << 30);  // addr hi, type=2

  v8u g1;
  g1.s0 = 0x00020000u;                      // data_size=2 (4B); no barrier/pad
  g1.s1 = (uint32_t)HID << 16;              // tensor_dim0 = 512 (low 16)
  g1.s2 = 0u;                               // dim0 hi, tensor_dim1 low16 = 0
  g1.s3 = ((uint32_t)HID << 16) | 1u;       // tile_dim0=512, tensor_dim1 hi=1 (65536 rows)
  g1.s4 = (uint32_t)ROWS_PER_CHUNK;         // tile_dim1 = #valid indices = 16
  g1.s5 = (uint32_t)HID;                    // tensor_dim0_stride = 512
  g1.s6 = 0u;
  g1.s7 = 0u;

  v4u g2, g3;                               // 16x 16-bit row indices
  g2.x = ridx[0]  | (ridx[1]  << 16);
  g2.y = ridx[2]  | (ridx[3]  << 16);
  g2.z = ridx[4]  | (ridx[5]  << 16);
  g2.w = ridx[6]  | (ridx[7]  << 16);
  g3.x = ridx[8]  | (ridx[9]  << 16);
  g3.y = ridx[10] | (ridx[11] << 16);
  g3.z = ridx[12] | (ridx[13] << 16);
  g3.w = ridx[14] | (ridx[15] << 16);

  asm volatile("tensor_load_to_lds %0, %1, %2, %3"
               :: "s"(g0), "s"(g1), "s"(g2), "s"(g3)
               : "memory");

  __builtin_amdgcn_s_wait_tensorcnt(0);     // gather tile resident in LDS

  // ---- D# for the contiguous store (normal mode, 2D 16x512 tile) --------
  const uint64_t go = (uint64_t)(uintptr_t)out + (uint64_t)row0 * ROW_BYTES;
  v4u s0;
  s0.x = 0x00000001u;                       // count=1, normal mode
  s0.y = lds;
  s0.z = (uint32_t)go;
  s0.w = ((uint32_t)(go >> 32) & 0x1FFFFFFu) | (2u << 30);

  v8u s1;
  s1.s0 = 0x00020000u;                      // data_size = 4B
  s1.s1 = (uint32_t)HID << 16;              // tensor_dim0 = 512
  s1.s2 = 0u;
  s1.s3 = ((uint32_t)HID << 16) | 1u;       // tile_dim0=512, tensor_dim1=65536
  s1.s4 = (uint32_t)ROWS_PER_CHUNK;         // tile_dim1 = 16 rows
  s1.s5 = (uint32_t)HID;                    // tensor_dim0_stride = 512
  s1.s6 = 0u;
  s1.s7 = 0u;

  asm volatile("tensor_store_from_lds %0, %1"
               :: "s"(s0), "s"(s1)
               : "memory");

  __builtin_amdgcn_s_wait_tensorcnt(0);     // drain before wave exit
}

// ---------------------------------------------------------------------------
extern "C" void kernel_launch(void* const* d_in, const int* in_sizes, int n_in,
                              void* d_out, int out_size, void* d_ws, size_t ws_size,
                              hipStream_t stream) {
  (void)in_sizes; (void)n_in; (void)out_size; (void)ws_size;

  const float* particles = (const float*)d_in[0];  // [P*B, H] f32
  const float* prob      = (const float*)d_in[1];  // [P*B, 1] f32 (log-weights)

  float* out_particles = (float*)d_out;                               // [P*B, H]
  float* out_prob      = out_particles + (size_t)PNUM * BATCH * HID;  // [P*B, 1]
  int*   srcrow        = (int*)d_ws;                                  // [P*B]

  // Steps 1+2+4 bookkeeping: tiny, one block per batch element.
  resample_kernel<<<BATCH, PNUM, 0, stream>>>(prob, srcrow, out_prob);

  // Step 3: 128 MB row gather via the Tensor Data Mover (gather mode).
  const int total_rows = PNUM * BATCH;                     // 65536
  gather_tdm_kernel<<<total_rows / ROWS_PER_CHUNK, 32, 0, stream>>>(
      particles, srcrow, out_particles);
}